// graphsage_71356586656058
// MI455X (gfx1250) — compile-verified
//
#include <hip/hip_runtime.h>

// ---- problem constants (from reference) ----
#define IN_F   128
#define H_F    256
#define N_CLS  47
#define BATCH  1024
#define NDST0  11264
#define E0_N   168960
#define E1_N   10240

typedef __attribute__((ext_vector_type(2))) float v2f;
typedef __attribute__((ext_vector_type(8))) float v8f;

// Async global->LDS copy of 16 bytes (gfx1250 GLOBAL_LOAD_ASYNC_TO_LDS_B128,
// tracked on ASYNCcnt). lds_off = low 32 bits of the generic LDS pointer
// (the flat LDS aperture takes those low bits as the LDS address).
__device__ __forceinline__ void async_copy_b128(uint32_t lds_off,
                                                const void* gptr) {
  uint64_t ga = (uint64_t)(uintptr_t)gptr;
  asm volatile("global_load_async_to_lds_b128 %0, %1, off"
               :: "v"(lds_off), "v"(ga)
               : "memory");
}
__device__ __forceinline__ void async_wait0() {
  asm volatile("s_wait_asynccnt 0x0" ::: "memory");
}

// 1/max(deg,1): deg is a small positive integer count, so a 1-ulp v_rcp_f32
// is exact enough and avoids the IEEE div_scale/div_fixup sequence.
__device__ __forceinline__ float inv_deg(float deg) {
  return __builtin_amdgcn_rcpf(fmaxf(deg, 1.0f));
}

// ---------------------------------------------------------------------------
// Zero-fill workspace accumulators (deterministic per call; graph-safe).
// ---------------------------------------------------------------------------
__global__ void zero_kernel(float* __restrict__ p, int n) {
  int i = blockIdx.x * blockDim.x + threadIdx.x;
  int stride = gridDim.x * blockDim.x;
  for (; i < n; i += stride) p[i] = 0.0f;
}

// ---------------------------------------------------------------------------
// Layer-0 edge aggregation: one wave32 per edge, 128 floats = float4/lane.
// h_in[s] = (s < B) ? x[s] : hist0[n_id0[s]]  (scatter into hist0 is dead:
// n_id0 is a permutation, so updated rows are never re-read).
// ---------------------------------------------------------------------------
__global__ void agg0_kernel(const float* __restrict__ x,
                            const float* __restrict__ hist0,
                            const int*   __restrict__ n_id0,
                            const int*   __restrict__ src0,
                            const int*   __restrict__ dst0,
                            float* __restrict__ neigh1,
                            float* __restrict__ deg0) {
  int e    = (int)((blockIdx.x * blockDim.x + threadIdx.x) >> 5);
  int lane = threadIdx.x & 31;
  if (e >= E0_N) return;
  int s = src0[e];
  int d = dst0[e];
  const float* row = (s < BATCH) ? (x + (size_t)s * IN_F)
                                 : (hist0 + (size_t)n_id0[s] * IN_F);
  float4 v = *(const float4*)(row + lane * 4);
  float* acc = neigh1 + (size_t)d * IN_F + lane * 4;
  atomicAdd(acc + 0, v.x);
  atomicAdd(acc + 1, v.y);
  atomicAdd(acc + 2, v.z);
  atomicAdd(acc + 3, v.w);
  if (lane == 0) atomicAdd(deg0 + d, 1.0f);
}

// ---------------------------------------------------------------------------
// Layer-1 fused GEMM: h = relu([h_dst | neigh1/deg] @ [W_self1; W_neigh1] + b1)
// M = NDST0, composite K = 256, N = 256. Block = 256 threads (8 waves); A-tile
// (16x256 f32, raw) staged via GLOBAL_LOAD_ASYNC_TO_LDS_B128; the 1/deg mean
// normalization is applied per-lane at A-fragment read (row = lane&15). Each
// wave computes two 16x16 tiles with V_WMMA_F32_16X16X4_F32 (2x32 k-steps).
// ---------------------------------------------------------------------------
__global__ void __launch_bounds__(256)
gemm1_kernel(const float* __restrict__ x, const float* __restrict__ hist0,
             const int* __restrict__ n_id0,
             const float* __restrict__ neigh1, const float* __restrict__ deg0,
             const float* __restrict__ Ws1, const float* __restrict__ Wn1,
             const float* __restrict__ b1, float* __restrict__ h) {
  __shared__ __align__(16) float As[16][264];   // 256 + 8 pad (bank skew)
  const int m0 = blockIdx.x * 16;
  const int t  = threadIdx.x;

  // ---- stage raw A' = [h_dst | neigh] tile into LDS (async copies) ----
  {
    int r    = t >> 4;        // 16 threads per row
    int cseg = t & 15;        // 16-float segment
    int m    = m0 + r;
    const float* src;
    if (cseg < 8) {           // k in [0,128): self features
      const float* row = (m < BATCH) ? (x + (size_t)m * IN_F)
                                     : (hist0 + (size_t)n_id0[m] * IN_F);
      src = row + cseg * 16;
    } else {                  // k in [128,256): raw neighbor sums
      src = neigh1 + (size_t)m * IN_F + (cseg - 8) * 16;
    }
    uint32_t lds = (uint32_t)(uintptr_t)&As[r][cseg * 16];
    #pragma unroll
    for (int j = 0; j < 4; ++j)
      async_copy_b128(lds + j * 16, src + j * 4);
    async_wait0();
  }
  __syncthreads();

  const int w    = t >> 5;
  const int lane = t & 31;
  const int arow = lane & 15;          // A fragment row / B fragment col
  const int koff = (lane >> 4) << 1;   // lanes 16-31 hold K+2,K+3
  const float* Arow = &As[arow][koff]; // lane-constant LDS base
  const float invd = inv_deg(deg0[m0 + arow]);

  #pragma unroll
  for (int i = 0; i < 2; ++i) {
    const int n0  = (w * 2 + i) * 16;
    const int col = n0 + arow;
    const float* B0 = Ws1 + (size_t)koff * H_F + col;  // self weights
    const float* B1 = Wn1 + (size_t)koff * H_F + col;  // neigh weights
    v8f acc = {};
    #pragma unroll 8
    for (int kk = 0; kk < IN_F; kk += 4) {             // self half
      v2f a, b;
      a.x = Arow[kk];
      a.y = Arow[kk + 1];
      b.x = B0[(size_t)kk * H_F];
      b.y = B0[(size_t)(kk + 1) * H_F];
      acc = __builtin_amdgcn_wmma_f32_16x16x4_f32(
          false, a, false, b, (short)0, acc, false, false);
    }
    #pragma unroll 8
    for (int kk = 0; kk < IN_F; kk += 4) {             // neighbor half (mean)
      v2f a, b;
      a.x = Arow[IN_F + kk] * invd;
      a.y = Arow[IN_F + kk + 1] * invd;
      b.x = B1[(size_t)kk * H_F];
      b.y = B1[(size_t)(kk + 1) * H_F];
      acc = __builtin_amdgcn_wmma_f32_16x16x4_f32(
          false, a, false, b, (short)0, acc, false, false);
    }
    const int mbase = m0 + ((lane >> 4) << 3);
    const float bias = b1[col];
    #pragma unroll
    for (int vv = 0; vv < 8; ++vv) {
      float val = acc[vv] + bias;
      h[(size_t)(mbase + vv) * H_F + col] = fmaxf(val, 0.0f);
    }
  }
}

// ---------------------------------------------------------------------------
// Layer-1 edge aggregation: one wave per edge, 256 floats = 2x float4/lane.
// h2[s] = (s < B) ? h[s] : hist1[n_id0[s]]   (hist1 scatter is dead too).
// ---------------------------------------------------------------------------
__global__ void agg1_kernel(const float* __restrict__ h,
                            const float* __restrict__ hist1,
                            const int*   __restrict__ n_id0,
                            const int*   __restrict__ src1,
                            const int*   __restrict__ dst1,
                            float* __restrict__ neigh2,
                            float* __restrict__ deg1) {
  int e    = (int)((blockIdx.x * blockDim.x + threadIdx.x) >> 5);
  int lane = threadIdx.x & 31;
  if (e >= E1_N) return;
  int s = src1[e];
  int d = dst1[e];
  const float* row = (s < BATCH) ? (h + (size_t)s * H_F)
                                 : (hist1 + (size_t)n_id0[s] * H_F);
  const float4* rp = (const float4*)(row + lane * 8);
  float4 v0 = rp[0], v1 = rp[1];
  float* acc = neigh2 + (size_t)d * H_F + lane * 8;
  atomicAdd(acc + 0, v0.x); atomicAdd(acc + 1, v0.y);
  atomicAdd(acc + 2, v0.z); atomicAdd(acc + 3, v0.w);
  atomicAdd(acc + 4, v1.x); atomicAdd(acc + 5, v1.y);
  atomicAdd(acc + 6, v1.z); atomicAdd(acc + 7, v1.w);
  if (lane == 0) atomicAdd(deg1 + d, 1.0f);
}

// ---------------------------------------------------------------------------
// Layer-2 fused GEMM: out = [h[:B] | neigh2/deg] @ [W_self2; W_neigh2] + b2
// M = 1024, composite K = 512, N = 47 (3 n-tiles, last partial). Columns
// >= 47 are clamped for addressing and masked to 0.0 in the B fragment, so
// loads are unconditional and WMMA runs with EXEC all-1s; stores guarded.
// Block = 128 threads (4 waves): waves 0-2 compute, all 4 stage LDS (async).
// ---------------------------------------------------------------------------
__global__ void __launch_bounds__(128)
gemm2_kernel(const float* __restrict__ h, const float* __restrict__ neigh2,
             const float* __restrict__ deg1,
             const float* __restrict__ Ws2, const float* __restrict__ Wn2,
             const float* __restrict__ b2, float* __restrict__ out) {
  __shared__ __align__(16) float As[16][520];   // 512 + 8 pad
  const int m0 = blockIdx.x * 16;
  const int t  = threadIdx.x;

  {
    int r   = t >> 3;        // 8 threads per row
    int seg = t & 7;         // 64-float segment
    int m   = m0 + r;
    const float* src = (seg < 4)
        ? (h + (size_t)m * H_F + seg * 64)                 // k in [0,256)
        : (neigh2 + (size_t)m * H_F + (seg - 4) * 64);     // raw neigh sums
    uint32_t lds = (uint32_t)(uintptr_t)&As[r][seg * 64];
    #pragma unroll
    for (int j = 0; j < 16; ++j)
      async_copy_b128(lds + j * 16, src + j * 4);
    async_wait0();
  }
  __syncthreads();

  const int w    = t >> 5;
  const int lane = t & 31;
  if (w < 3) {
    const int arow = lane & 15;
    const int koff = (lane >> 4) << 1;
    const int n0   = w * 16;
    const int col  = n0 + arow;
    const bool col_ok = (col < N_CLS);
    const int  colc   = col_ok ? col : (N_CLS - 1);   // clamp: in-bounds addr
    const float mask  = col_ok ? 1.0f : 0.0f;         // zero padded B columns
    const float* Arow = &As[arow][koff];
    const float invd  = inv_deg(deg1[m0 + arow]);
    const float* B0 = Ws2 + (size_t)koff * N_CLS + colc;
    const float* B1 = Wn2 + (size_t)koff * N_CLS + colc;
    v8f acc = {};
    #pragma unroll 8
    for (int kk = 0; kk < H_F; kk += 4) {              // self half
      v2f a, b;
      a.x = Arow[kk];
      a.y = Arow[kk + 1];
      b.x = B0[(size_t)kk * N_CLS] * mask;
      b.y = B0[(size_t)(kk + 1) * N_CLS] * mask;
      acc = __builtin_amdgcn_wmma_f32_16x16x4_f32(
          false, a, false, b, (short)0, acc, false, false);
    }
    #pragma unroll 8
    for (int kk = 0; kk < H_F; kk += 4) {              // neighbor half (mean)
      v2f a, b;
      a.x = Arow[H_F + kk] * invd;
      a.y = Arow[H_F + kk + 1] * invd;
      b.x = B1[(size_t)kk * N_CLS] * mask;
      b.y = B1[(size_t)(kk + 1) * N_CLS] * mask;
      acc = __builtin_amdgcn_wmma_f32_16x16x4_f32(
          false, a, false, b, (short)0, acc, false, false);
    }
    if (col_ok) {
      const int mbase = m0 + ((lane >> 4) << 3);
      const float bias = b2[col];
      #pragma unroll
      for (int vv = 0; vv < 8; ++vv)
        out[(size_t)(mbase + vv) * N_CLS + col] = acc[vv] + bias;
    }
  }
}

// ---------------------------------------------------------------------------
extern "C" void kernel_launch(void* const* d_in, const int* in_sizes, int n_in,
                              void* d_out, int out_size, void* d_ws,
                              size_t ws_size, hipStream_t stream) {
  const float* x     = (const float*)d_in[0];
  const float* hist0 = (const float*)d_in[1];
  const float* hist1 = (const float*)d_in[2];
  const float* Ws1   = (const float*)d_in[3];
  const float* Wn1   = (const float*)d_in[4];
  const float* b1    = (const float*)d_in[5];
  const float* Ws2   = (const float*)d_in[6];
  const float* Wn2   = (const float*)d_in[7];
  const float* b2    = (const float*)d_in[8];
  const int* src0    = (const int*)d_in[9];
  const int* dst0    = (const int*)d_in[10];
  const int* src1    = (const int*)d_in[11];
  const int* dst1    = (const int*)d_in[12];
  const int* n_id0   = (const int*)d_in[13];
  float* out = (float*)d_out;

  // workspace layout (floats): neigh1 | deg0 | neigh2 | deg1 | h
  float* ws     = (float*)d_ws;
  float* neigh1 = ws;
  float* deg0   = neigh1 + (size_t)NDST0 * IN_F;
  float* neigh2 = deg0 + NDST0;
  float* deg1   = neigh2 + (size_t)BATCH * H_F;
  float* h      = deg1 + BATCH;

  const int nzero = NDST0 * IN_F + NDST0 + BATCH * H_F + BATCH;
  zero_kernel<<<1024, 256, 0, stream>>>(ws, nzero);

  agg0_kernel<<<E0_N / 8, 256, 0, stream>>>(x, hist0, n_id0, src0, dst0,
                                            neigh1, deg0);
  gemm1_kernel<<<NDST0 / 16, 256, 0, stream>>>(x, hist0, n_id0, neigh1, deg0,
                                               Ws1, Wn1, b1, h);
  agg1_kernel<<<E1_N / 8, 256, 0, stream>>>(h, hist1, n_id0, src1, dst1,
                                            neigh2, deg1);
  gemm2_kernel<<<BATCH / 16, 128, 0, stream>>>(h, neigh2, deg1, Ws2, Wn2, b2,
                                               out);
}